// GCNLayer_1090921693858
// MI455X (gfx1250) — compile-verified
//
#include <hip/hip_runtime.h>
#include <hip/hip_bf16.h>

typedef __attribute__((ext_vector_type(2))) float v2f;
typedef __attribute__((ext_vector_type(8))) float v8f;

#define NNODES 50000
#define NEDGES 800000
#define IN_CH  256
#define HID_CH 128

// ---------------- utility: fill a float buffer with a constant ----------------
__global__ void fill_kernel(float* __restrict__ p, float v, int n) {
    int i = blockIdx.x * blockDim.x + threadIdx.x;
    int stride = gridDim.x * blockDim.x;
    for (; i < n; i += stride) p[i] = v;
}

// ---------------- degree: deg[dst] += 1 (deg pre-filled with 1.0 = self loop) --
__global__ void degree_kernel(const int* __restrict__ dst, float* __restrict__ deg, int E) {
    int e = blockIdx.x * blockDim.x + threadIdx.x;
    if (e < E) atomicAdd(&deg[dst[e]], 1.0f);
}

// ---------------- H[M x 128] = X[M x K] @ W[K x 128] via V_WMMA_F32_16X16X4_F32
// One wave per 16-row strip; 8 column tiles of 16; K stepped by 4.
// A-frag (16x4 f32): lanes 0-15 -> M=0..15 hold K=k0+0 (v0), k0+1 (v1);
//                    lanes 16-31 hold K=k0+2 (v0), k0+3 (v1).
// B-frag (4x16 f32): mirrored — lane half selects K pair, lane&15 selects N.
// C/D (16x16 f32): VGPR j, lane L -> M = j + 8*(L>=16), N = L&15.
__global__ void gemm_wmma_f32(const float* __restrict__ X, const float* __restrict__ W,
                              float* __restrict__ H, int M, int K) {
    const int lane   = threadIdx.x & 31;
    const int wid    = threadIdx.x >> 5;
    const int tile   = blockIdx.x * 4 + wid;   // 16-row strip index
    const int mtiles = M >> 4;
    if (tile >= mtiles) return;                // uniform per wave: EXEC stays all-ones

    const int half = lane >> 4;                // 0 or 1
    const int r    = lane & 15;
    const int row  = tile * 16 + r;
    const float* xrow = X + (size_t)row * K;

    for (int nt = 0; nt < 8; ++nt) {
        const int col = nt * 16 + r;
        v8f acc = {};
        for (int k0 = 0; k0 < K; k0 += 4) {
            v2f a, b;
            a.x = xrow[k0 + 2 * half];
            a.y = xrow[k0 + 2 * half + 1];
            b.x = W[(size_t)(k0 + 2 * half)     * HID_CH + col];
            b.y = W[(size_t)(k0 + 2 * half + 1) * HID_CH + col];
            acc = __builtin_amdgcn_wmma_f32_16x16x4_f32(
                /*neg_a=*/false, a, /*neg_b=*/false, b,
                /*c_mod=*/(short)0, acc, /*reuse_a=*/false, /*reuse_b=*/false);
        }
#pragma unroll
        for (int j = 0; j < 8; ++j) {
            H[(size_t)(tile * 16 + j + 8 * half) * HID_CH + col] = acc[j];
        }
    }
}

// ---------------- edge aggregation: agg[dst] += norm * h[src] -----------------
// 32 threads per edge, 4 channels each (C = 128). h is L2-resident (25.6 MB).
__global__ void edge_agg_kernel(const int* __restrict__ src, const int* __restrict__ dst,
                                const float* __restrict__ deg,
                                const float* __restrict__ h, float* __restrict__ agg) {
    int t = blockIdx.x * blockDim.x + threadIdx.x;
    int e = t >> 5;
    if (e >= NEDGES) return;
    int c4 = (t & 31) << 2;
    int s = src[e];
    int d = dst[e];
    float norm = rsqrtf(deg[s]) * rsqrtf(deg[d]);
    const float4 hv = *(const float4*)(h + (size_t)s * HID_CH + c4);
    float* ap = agg + (size_t)d * HID_CH + c4;
    atomicAdd(ap + 0, norm * hv.x);
    atomicAdd(ap + 1, norm * hv.y);
    atomicAdd(ap + 2, norm * hv.z);
    atomicAdd(ap + 3, norm * hv.w);
}

// ---------------- finalize: out = agg + h/deg + bias (+ optional LeakyReLU) ---
__global__ void finalize_kernel(const float* __restrict__ agg, const float* __restrict__ h,
                                const float* __restrict__ deg, const float* __restrict__ bias,
                                float* __restrict__ out, int leaky) {
    int t = blockIdx.x * blockDim.x + threadIdx.x;
    if (t >= NNODES * HID_CH) return;
    int rowd = t >> 7;      // /128
    int c    = t & 127;
    float v = agg[t] + h[t] * (1.0f / deg[rowd]) + bias[c];
    if (leaky) v = (v >= 0.0f) ? v : 0.01f * v;
    out[t] = v;
}

extern "C" void kernel_launch(void* const* d_in, const int* in_sizes, int n_in,
                              void* d_out, int out_size, void* d_ws, size_t ws_size,
                              hipStream_t stream) {
    const float* x  = (const float*)d_in[0];
    const int*   ei = (const int*)d_in[1];     // [2, E] flattened: row 0 = src, row 1 = dst
    const float* W1 = (const float*)d_in[2];
    const float* b1 = (const float*)d_in[3];
    const float* W2 = (const float*)d_in[4];
    const float* b2 = (const float*)d_in[5];
    const int* src = ei;
    const int* dst = ei + NEDGES;

    float* ws  = (float*)d_ws;
    float* deg = ws;                                   // 50000 floats (pad to 51200)
    float* h   = ws + 51200;                           // 50000 x 128
    float* agg = h + (size_t)NNODES * HID_CH;          // 50000 x 128
    float* act = agg + (size_t)NNODES * HID_CH;        // 50000 x 128
    float* out = (float*)d_out;

    const int featN = NNODES * HID_CH;                 // 6.4M
    const int mtiles = NNODES / 16;                    // 3125
    const dim3 gemmGrid((mtiles + 3) / 4);             // 4 waves (strips) per 128-thr block
    const dim3 gemmBlock(128);
    const int edgeBlocks = (NEDGES * 32 + 255) / 256;  // 100000
    const int featBlocks = (featN + 255) / 256;

    // degree (shared by both layers)
    fill_kernel<<<256, 256, 0, stream>>>(deg, 1.0f, NNODES);
    degree_kernel<<<(NEDGES + 255) / 256, 256, 0, stream>>>(dst, deg, NEDGES);

    // ---- layer 1: h = x @ W1 ; aggregate ; finalize + LeakyReLU -> act ----
    fill_kernel<<<4096, 256, 0, stream>>>(agg, 0.0f, featN);
    gemm_wmma_f32<<<gemmGrid, gemmBlock, 0, stream>>>(x, W1, h, NNODES, IN_CH);
    edge_agg_kernel<<<edgeBlocks, 256, 0, stream>>>(src, dst, deg, h, agg);
    finalize_kernel<<<featBlocks, 256, 0, stream>>>(agg, h, deg, b1, act, 1);

    // ---- layer 2: h = act @ W2 ; aggregate ; finalize -> d_out ----
    fill_kernel<<<4096, 256, 0, stream>>>(agg, 0.0f, featN);
    gemm_wmma_f32<<<gemmGrid, gemmBlock, 0, stream>>>(act, W2, h, NNODES, HID_CH);
    edge_agg_kernel<<<edgeBlocks, 256, 0, stream>>>(src, dst, deg, h, agg);
    finalize_kernel<<<featBlocks, 256, 0, stream>>>(agg, h, deg, b2, out, 0);
}